// HungarianMatcher_15006615732337
// MI455X (gfx1250) — compile-verified
//
#include <hip/hip_runtime.h>
#include <hip/hip_bf16.h>
#include <math.h>

#define BB 64
#define NN 900
#define CC 128
#define MM 64
#define KK 8
#define NT 57            /* ceil(900/16) */
#define COSTW_CLASS 1.0f
#define COSTW_BBOX  5.0f
#define LSA_INF 1.0e9f

typedef float v2f __attribute__((ext_vector_type(2)));
typedef float v8f __attribute__((ext_vector_type(8)));

// ---------------------------------------------------------------------------
// Cost kernel: one wave32 computes a 16(row)x64(col) tile of C[b,n,m].
//   class part: online softmax (two half-lanes per row, combined via shfl_xor)
//   box part  : ||c-b|| = sqrt(|c|^2 - 2*c.b + |b|^2), c.b via WMMA f32 16x16x4
// ---------------------------------------------------------------------------
__global__ __launch_bounds__(256)
void cost_kernel(const float* __restrict__ logits,
                 const float* __restrict__ corners,
                 const int*   __restrict__ labels,
                 const float* __restrict__ boxes,
                 float* __restrict__ Cout)
{
  const int wave = blockIdx.x * 8 + (threadIdx.x >> 5);
  if (wave >= BB * NT) return;                 // uniform per wave
  const int lane = threadIdx.x & 31;
  const int b    = wave / NT;
  const int tile = wave % NT;
  const int hi   = lane >> 4;                  // 0: K=0,1  1: K=2,3
  const int r    = lane & 15;                  // row (A) / col (B) in tile
  const int n    = tile * 16 + r;
  const bool valid = (n < NN);

  // ---- centroid of the 8 corners for row r
  float cx = 0.f, cy = 0.f, cz = 0.f;
  if (valid) {
    const float* cp = corners + (((size_t)b * NN + n) * KK) * 3;
    #pragma unroll
    for (int k = 0; k < KK; ++k) {
      cx += cp[k * 3 + 0]; cy += cp[k * 3 + 1]; cz += cp[k * 3 + 2];
    }
    cx *= 0.125f; cy *= 0.125f; cz *= 0.125f;
  }
  const float cn2 = cx * cx + cy * cy + cz * cz;

  // ---- online softmax stats for row r: this half-lane scans 64 of 128 logits
  float mloc = -3.0e38f, sloc = 0.f;
  if (valid) {
    const float* lp = logits + ((size_t)b * NN + n) * CC + hi * 64;
    for (int c = 0; c < 64; ++c) {
      float x  = lp[c];
      float mn = fmaxf(mloc, x);
      sloc = sloc * __expf(mloc - mn) + __expf(x - mn);
      mloc = mn;
    }
  }
  {
    float mo = __shfl_xor(mloc, 16, 32);
    float so = __shfl_xor(sloc, 16, 32);
    float mt = fmaxf(mloc, mo);
    float st = sloc * __expf(mloc - mt) + so * __expf(mo - mt);
    mloc = mt; sloc = st;
  }
  if (!valid) { mloc = 0.f; sloc = 1.f; }
  const float sinv = 1.0f / sloc;              // one divide per lane, not per cell

  // ---- per-lane copies of the 8 rows this lane owns in the C/D fragment
  float rmax[8], rinv[8], rcn2[8];
  #pragma unroll
  for (int i = 0; i < 8; ++i) {
    int src = i + 8 * hi;                      // rows 0-7 (lo half) / 8-15 (hi)
    rmax[i] = __shfl(mloc, src, 32);
    rinv[i] = __shfl(sinv, src, 32);
    rcn2[i] = __shfl(cn2,  src, 32);
  }

  // ---- A fragment: 16x4 f32, lanes 0-15 K={0,1}, lanes 16-31 K={2,3}
  v2f amat;
  amat.x = hi ? cz : cx;
  amat.y = hi ? 0.f : cy;

  const float* logits_b = logits + (size_t)b * NN * CC;

  #pragma unroll
  for (int mc = 0; mc < 4; ++mc) {
    const int mcol = mc * 16 + r;
    const float* bp = boxes + ((size_t)b * MM + mcol) * 7;
    const float bx = bp[0], by = bp[1], bz = bp[2];
    const float bn2 = bx * bx + by * by + bz * bz;
    const int lab = labels[b * MM + mcol];

    // B fragment: 4x16 f32, VGPR0 lanes0-15 K=0 / lanes16-31 K=2, VGPR1 K=1/K=3
    v2f bmat;
    bmat.x = hi ? bz : bx;
    bmat.y = hi ? 0.f : by;

    v8f acc = {};
    acc = __builtin_amdgcn_wmma_f32_16x16x4_f32(false, amat, false, bmat,
                                                (short)0, acc, false, false);

    #pragma unroll
    for (int i = 0; i < 8; ++i) {
      const int nrow = tile * 16 + i + 8 * hi;
      if (nrow < NN) {
        float dot = acc[i];
        float d2  = fmaxf(rcn2[i] - 2.f * dot + bn2, 0.f);
        float bc  = __builtin_amdgcn_sqrtf(d2);      // single v_sqrt_f32
        float lg  = logits_b[(size_t)nrow * CC + lab];
        float cl  = -__expf(lg - rmax[i]) * rinv[i]; // mul, not divide
        Cout[((size_t)b * NN + nrow) * MM + mcol] =
            COSTW_CLASS * cl + COSTW_BBOX * bc;
      }
    }
  }
}

// ---------------------------------------------------------------------------
// Jonker-Volgenant LSA: one 256-thread workgroup per batch.
// cost[m][n] = C[b][n][m]; column-parallel inner step + LDS tree argmin.
// ---------------------------------------------------------------------------
__global__ __launch_bounds__(256)
void lsa_kernel(const float* __restrict__ Cmat,
                int* __restrict__ pred, int* __restrict__ tgt)
{
  const int b   = blockIdx.x;
  const int tid = threadIdx.x;
  const float* Cb = Cmat + (size_t)b * NN * MM;

  __shared__ float v[NN + 1];
  __shared__ float minv[NN + 1];
  __shared__ int   way[NN + 1];
  __shared__ int   p[NN + 1];
  __shared__ unsigned char used[NN + 1];
  __shared__ float u[MM + 1];
  __shared__ float red_val[256];
  __shared__ int   red_idx[256];
  __shared__ int   s_j0;

  for (int j = tid; j <= NN; j += 256) { v[j] = 0.f; p[j] = 0; }
  for (int j = tid; j <= MM; j += 256) u[j] = 0.f;
  __syncthreads();

  for (int i1 = 1; i1 <= MM; ++i1) {
    if (tid == 0) { p[0] = i1; s_j0 = 0; }
    for (int j = tid; j <= NN; j += 256) {
      minv[j] = LSA_INF; way[j] = 0; used[j] = 0;
    }
    __syncthreads();

    while (true) {
      const int j0 = s_j0;                 // uniform
      const int i0 = p[j0];                // uniform (p stable inside loop)
      if (i0 == 0) break;                  // uniform exit
      if (tid == 0) used[j0] = 1;
      const float ui0 = u[i0];
      __syncthreads();

      // scan: update minv/way, local masked argmin
      float lmin = LSA_INF; int lidx = 0;
      const int row = i0 - 1;
      for (int j = tid; j <= NN; j += 256) {
        if (!used[j]) {
          float aij = (j == 0) ? 0.f : Cb[(size_t)(j - 1) * MM + row];
          float cur = aij - ui0 - v[j];
          float mv  = minv[j];
          if (cur < mv) { mv = cur; minv[j] = cur; way[j] = j0; }
          if (mv < lmin) { lmin = mv; lidx = j; }
        }
      }
      red_val[tid] = lmin; red_idx[tid] = lidx;
      __syncthreads();
      for (int s = 128; s > 0; s >>= 1) {
        if (tid < s) {
          if (red_val[tid + s] < red_val[tid]) {
            red_val[tid] = red_val[tid + s];
            red_idx[tid] = red_idx[tid + s];
          }
        }
        __syncthreads();
      }
      const float delta = red_val[0];
      const int   j1    = red_idx[0];
      __syncthreads();

      // dual updates: used columns have distinct nonzero p[j] -> race-free
      for (int j = tid; j <= NN; j += 256) {
        if (used[j]) {
          u[p[j]] += delta;
          v[j]    -= delta;
        } else {
          minv[j] -= delta;
        }
      }
      if (tid == 0) s_j0 = j1;
      __syncthreads();
    }

    if (tid == 0) {                        // backtrack augmenting path
      int j0 = s_j0;
      while (j0 != 0) { int j1 = way[j0]; p[j0] = p[j1]; j0 = j1; }
    }
    __syncthreads();
  }

  // outputs: res scatter (rows unique), tgt = iota
  for (int m2 = tid; m2 < MM; m2 += 256) pred[(size_t)b * MM + m2] = 0;
  __syncthreads();
  for (int nn = tid; nn < NN; nn += 256) {
    int rowv = p[nn + 1];
    if (rowv > 0) pred[(size_t)b * MM + (rowv - 1)] = nn;
  }
  for (int m2 = tid; m2 < MM; m2 += 256) tgt[(size_t)b * MM + m2] = m2;
}

// ---------------------------------------------------------------------------
extern "C" void kernel_launch(void* const* d_in, const int* in_sizes, int n_in,
                              void* d_out, int out_size, void* d_ws, size_t ws_size,
                              hipStream_t stream) {
  const float* logits  = (const float*)d_in[0];   // [B,N,128] f32
  const float* corners = (const float*)d_in[1];   // [B,N,8,3] f32
  const int*   labels  = (const int*)  d_in[2];   // [B,M] i32
  const float* boxes   = (const float*)d_in[3];   // [B,M,7] f32

  float* Cout = (float*)d_out;                              // [B,N,M] f32
  int*   pred = (int*)(Cout + (size_t)BB * NN * MM);        // [B,M] i32 bits
  int*   tgt  = pred + (size_t)BB * MM;                     // [B,M] i32 bits

  const int waves  = BB * NT;            // 3648 wave-tiles
  const int blocks = (waves + 7) / 8;    // 8 waves / 256-thread block

  cost_kernel<<<blocks, 256, 0, stream>>>(logits, corners, labels, boxes, Cout);
  lsa_kernel<<<BB, 256, 0, stream>>>(Cout, pred, tgt);
}